// PolyIoULoss_75153337745755
// MI455X (gfx1250) — compile-verified
//
#include <hip/hip_runtime.h>
#include <math.h>

// ext-vector types for WMMA f32 16x16x4: A=16x4 f32 (2 VGPR), B=4x16 f32 (2 VGPR), C/D=16x16 f32 (8 VGPR)
typedef float v2f __attribute__((ext_vector_type(2)));
typedef float v8f __attribute__((ext_vector_type(8)));

#define TPB 256

// ---------------------------------------------------------------------------
// Stage 1: per-pair rotated-box IoU loss via Sutherland-Hodgman clipping.
// One pair per lane (wave32). Clip vertex buffers live in LDS (dynamic
// indexing; conflict-free layout i*TPB+tid). Deterministic LDS tree reduce
// -> one partial per block.
// ---------------------------------------------------------------------------
__global__ __launch_bounds__(TPB) void obb_iou_loss_kernel(
    const float* __restrict__ pred, const float* __restrict__ targ,
    float* __restrict__ partial, int n, int perThread)
{
    __shared__ float bx0[8 * TPB], by0[8 * TPB];
    __shared__ float bx1[8 * TPB], by1[8 * TPB];
    __shared__ float red[TPB];

    const int tid = threadIdx.x;
    float acc = 0.0f;

    for (int k = 0; k < perThread; ++k) {
        const int idx = (blockIdx.x * perThread + k) * TPB + tid;
        if (idx < n) {
            // Prefetch next chunk (gfx1250 global_prefetch_b8).
            if (idx + TPB < n) {
                __builtin_prefetch(pred + 5 * (size_t)(idx + TPB), 0, 0);
                __builtin_prefetch(targ + 5 * (size_t)(idx + TPB), 0, 0);
            }

            const float* P = pred + 5 * (size_t)idx;
            const float* T = targ + 5 * (size_t)idx;
            const float pcx = P[0], pcy = P[1], pw = P[2], ph = P[3], pth = P[4];
            const float tcx = T[0], tcy = T[1], tw = T[2], th2 = T[3], tth = T[4];

            float psn, pcs, tsn, tcs;
            __sincosf(pth, &psn, &pcs);
            __sincosf(tth, &tsn, &tcs);

            const float pdx = 0.5f * pw, pdy = 0.5f * ph;
            const float tdx = 0.5f * tw, tdy = 0.5f * th2;

            // Local corners CCW: (-dx,-dy),(dx,-dy),(dx,dy),(-dx,dy)
            // world = [lx*c + ly*s + cx, -lx*s + ly*c + cy]  (matches einsum in ref)
            float txv[4], tyv[4];
            {
                const float lx[4] = {-tdx, tdx, tdx, -tdx};
                const float ly[4] = {-tdy, -tdy, tdy, tdy};
#pragma unroll
                for (int i = 0; i < 4; ++i) {
                    txv[i] = fmaf(lx[i], tcs, fmaf(ly[i], tsn, tcx));
                    tyv[i] = fmaf(-lx[i], tsn, fmaf(ly[i], tcs, tcy));
                }
            }
            {
                const float lx[4] = {-pdx, pdx, pdx, -pdx};
                const float ly[4] = {-pdy, -pdy, pdy, pdy};
#pragma unroll
                for (int i = 0; i < 4; ++i) {
                    bx0[i * TPB + tid] = fmaf(lx[i], pcs, fmaf(ly[i], psn, pcx));
                    by0[i * TPB + tid] = fmaf(-lx[i], psn, fmaf(ly[i], pcs, pcy));
                }
            }

            // Clip P by the 4 CCW edges of T (keep left side: cross >= 0).
            float* curx = bx0; float* cury = by0;
            float* outx = bx1; float* outy = by1;
            int cnt = 4;
#pragma unroll
            for (int e = 0; e < 4; ++e) {
                const float ax = txv[e], ay = tyv[e];
                const float ex = txv[(e + 1) & 3] - ax;
                const float ey = tyv[(e + 1) & 3] - ay;
                int m = 0;
                for (int i = 0; i < cnt; ++i) {
                    const int j = (i + 1 == cnt) ? 0 : i + 1;
                    const float x0 = curx[i * TPB + tid], y0 = cury[i * TPB + tid];
                    const float x1 = curx[j * TPB + tid], y1 = cury[j * TPB + tid];
                    const float d0 = ex * (y0 - ay) - ey * (x0 - ax);
                    const float d1 = ex * (y1 - ay) - ey * (x1 - ax);
                    const bool in0 = (d0 >= 0.0f);
                    const bool in1 = (d1 >= 0.0f);
                    if (in0) { outx[m * TPB + tid] = x0; outy[m * TPB + tid] = y0; ++m; }
                    if (in0 != in1) {
                        const float t = d0 / (d0 - d1);
                        outx[m * TPB + tid] = fmaf(t, x1 - x0, x0);
                        outy[m * TPB + tid] = fmaf(t, y1 - y0, y0);
                        ++m;
                    }
                }
                cnt = m;
                float* s;
                s = curx; curx = outx; outx = s;
                s = cury; cury = outy; outy = s;
            }

            // Shoelace area of the clipped polygon.
            float area2 = 0.0f;
            for (int i = 0; i < cnt; ++i) {
                const int j = (i + 1 == cnt) ? 0 : i + 1;
                const float x0 = curx[i * TPB + tid], y0 = cury[i * TPB + tid];
                const float x1 = curx[j * TPB + tid], y1 = cury[j * TPB + tid];
                area2 = fmaf(x0, y1, area2) - x1 * y0;
            }
            const float inter = 0.5f * fabsf(area2);
            const float a1 = pw * ph;
            const float a2 = tw * th2;
            const float iou = fmaxf(inter / (a1 + a2 - inter + 1e-6f), 1e-6f);
            acc += -__logf(iou);
        }
    }

    // Deterministic block tree-reduction.
    red[tid] = acc;
    __syncthreads();
    for (int s = TPB / 2; s > 0; s >>= 1) {
        if (tid < s) red[tid] += red[tid + s];
        __syncthreads();
    }
    if (tid == 0) partial[blockIdx.x] = red[0];
}

// ---------------------------------------------------------------------------
// Stage 2: single-wave deterministic reduction of block partials.
// Final 32-lane sum is done with V_WMMA_F32_16X16X4_F32:
//   A (16x4): lane acc at K=0 (lanes 0-15 -> rows 0-15) and K=2 (lanes 16-31),
//   B (4x16) = ones  =>  D[m][n] = acc[m] + acc[m+16]  (replicated over n).
// Sum the 8 D components per lane, add lane^16 partner, lane 0 writes mean.
// ---------------------------------------------------------------------------
__global__ __launch_bounds__(32) void obb_reduce_kernel(
    const float* __restrict__ partial, float* __restrict__ out,
    int numPartial, float invN)
{
    const int lane = threadIdx.x;
    float acc = 0.0f;
    for (int i = lane; i < numPartial; i += 32) acc += partial[i];

    v2f a; a.x = acc;  a.y = 0.0f;
    v2f b; b.x = 1.0f; b.y = 1.0f;
    v8f c = {};
    // (neg_a, A, neg_b, B, c_mod, C, reuse_a, reuse_b)
    c = __builtin_amdgcn_wmma_f32_16x16x4_f32(false, a, false, b, (short)0, c,
                                              false, false);
    float s = c[0] + c[1] + c[2] + c[3] + c[4] + c[5] + c[6] + c[7];
    s += __shfl_xor(s, 16, 32);
    if (lane == 0) out[0] = s * invN;
}

extern "C" void kernel_launch(void* const* d_in, const int* in_sizes, int n_in,
                              void* d_out, int out_size, void* d_ws, size_t ws_size,
                              hipStream_t stream) {
    const float* pred = (const float*)d_in[0];
    const float* targ = (const float*)d_in[1];
    float* out = (float*)d_out;
    float* partial = (float*)d_ws;

    const int n = in_sizes[0] / 5;   // (N,5) boxes
    const int threads = TPB;

    long maxBlocks = (long)(ws_size / sizeof(float));
    if (maxBlocks > 2048) maxBlocks = 2048;
    if (maxBlocks < 1) maxBlocks = 1;
    long want = ((long)n + threads - 1) / threads;
    int blocks = (int)(want < maxBlocks ? want : maxBlocks);
    if (blocks < 1) blocks = 1;
    int perThread = (int)(((long)n + (long)blocks * threads - 1) /
                          ((long)blocks * threads));

    obb_iou_loss_kernel<<<blocks, threads, 0, stream>>>(pred, targ, partial, n,
                                                        perThread);
    obb_reduce_kernel<<<1, 32, 0, stream>>>(partial, out, blocks,
                                            1.0f / (float)n);
}